// GCN_G_42760694399685
// MI455X (gfx1250) — compile-verified
//
#include <hip/hip_runtime.h>
#include <hip/hip_bf16.h>
#include <stdint.h>

typedef float v2f __attribute__((ext_vector_type(2)));
typedef float v8f __attribute__((ext_vector_type(8)));
typedef unsigned int v4u __attribute__((ext_vector_type(4)));
typedef int v4i __attribute__((ext_vector_type(4)));
typedef int v8i __attribute__((ext_vector_type(8)));

#define DF 128        // feature dim == hidden dim
#define A_STRIDE 132  // 128 floats + 4-dword TDM pad per row (bank-conflict-free fragments)

// ---------------- degree / normalization ----------------
__global__ void k_deg_init(float* deg, int n) {
  int i = blockIdx.x * blockDim.x + threadIdx.x;
  if (i < n) deg[i] = 1.0f;  // self-loop contribution
}

__global__ void k_deg_count(const long long* __restrict__ ei, float* deg, int E) {
  int e = blockIdx.x * blockDim.x + threadIdx.x;
  if (e < E) atomicAdd(&deg[(int)ei[(size_t)E + e]], 1.0f);
}

__global__ void k_dinv(float* deg, int n) {
  int i = blockIdx.x * blockDim.x + threadIdx.x;
  if (i < n) deg[i] = rsqrtf(deg[i]);  // deg >= 1 always (self-loops)
}

__global__ void k_norm(const long long* __restrict__ ei, const float* __restrict__ dinv,
                       float* __restrict__ norm, int E) {
  int e = blockIdx.x * blockDim.x + threadIdx.x;
  if (e < E) norm[e] = dinv[(int)ei[e]] * dinv[(int)ei[(size_t)E + e]];
}

// ---------------- dense GEMM: TDM (A tile -> LDS) + fp32 WMMA ----------------
// T[m, n] = sum_k X[m, k] * W[k, n]   (X: nrows x 128, W: 128 x 128, row-major)
// Block = 256 threads = 8 waves; block covers 16 rows x 128 cols (one 16x16 tile/wave).
// Wave 0 DMAs the shared 16x128 A-tile into LDS via the Tensor Data Mover, with the
// TDM pad feature inserting 4 dwords per 128-dword row (row stride 132 floats).
__global__ void k_gemm(const float* __restrict__ X, const float* __restrict__ W,
                       float* __restrict__ T, int nrows) {
  __shared__ float As[16 * A_STRIDE];

  const int wave = threadIdx.x >> 5;
  const int lane = threadIdx.x & 31;
  const int half = lane >> 4;   // 0: lanes 0-15, 1: lanes 16-31
  const int l16  = lane & 15;
  const int m0 = blockIdx.x * 16;
  const int n0 = wave * 16;

  if (threadIdx.x < 32) {
    // ---- Tensor DMA descriptor (CDNA5 ISA) ----
    unsigned long long ga = (unsigned long long)(uintptr_t)(X + (size_t)m0 * DF);
    unsigned int lds = (unsigned int)(uintptr_t)(&As[0]);
    v4u g0;
    g0.x = 1u;                                                      // count=1 (valid user D#)
    g0.y = lds;                                                     // lds_addr
    g0.z = (unsigned int)ga;                                        // global_addr[31:0]
    g0.w = (unsigned int)((ga >> 32) & 0x01FFFFFFu) | (2u << 30);   // global_addr[56:32] | type=2
    v8i g1;
    // data_size=4B (2), pad_enable=1, pad_interval=6 (128 dwords), pad_amount=3 (4 dwords)
    g1[0] = (int)((2u << 16) | (1u << 20) | (6u << 22) | (3u << 25));
    g1[1] = (int)(((unsigned)DF & 0xFFFFu) << 16);                  // tensor_dim0 lo16 -> bits[63:48]
    g1[2] = (int)((((unsigned)DF >> 16) & 0xFFFFu) |                // tensor_dim0 hi16
                  (((unsigned)nrows & 0xFFFFu) << 16));             // tensor_dim1 lo16
    g1[3] = (int)((((unsigned)nrows >> 16) & 0xFFFFu) |             // tensor_dim1 hi16
                  ((unsigned)DF << 16));                            // tile_dim0 = 128
    g1[4] = 16;                                                     // tile_dim1 = 16, tile_dim2 = 0
    g1[5] = DF;                                                     // tensor_dim0_stride lo32 = 128
    g1[6] = 0;                                                      // stride hi16 | dim1_stride lo16
    g1[7] = 0;
    v4i gz = {};
#if defined(__clang_major__) && __clang_major__ >= 23
    v8i gz8 = {};
    __builtin_amdgcn_tensor_load_to_lds(g0, g1, gz, gz, gz8, 0);
#else
    __builtin_amdgcn_tensor_load_to_lds(g0, g1, gz, gz, 0);
#endif
    __builtin_amdgcn_s_wait_tensorcnt(0);
  }
  __syncthreads();

  // A fragment (16x4 f32): lanes 0-15 -> K = k+0,k+1 ; lanes 16-31 -> K = k+2,k+3 (from LDS)
  const float* a_lds = &As[(size_t)l16 * A_STRIDE + 2 * half];
  // B fragment (4x16 f32): mirrored layout, N = n0 + l16 striped across lanes (from global/L2)
  const float* b_ptr = W + (size_t)(2 * half) * DF + n0 + l16;

  v8f c = {};
#pragma unroll
  for (int k = 0; k < DF; k += 4) {
    v2f a = *(const v2f*)(a_lds + k);
    v2f b;
    b.x = b_ptr[(size_t)k * DF];
    b.y = b_ptr[(size_t)(k + 1) * DF];
    c = __builtin_amdgcn_wmma_f32_16x16x4_f32(false, a, false, b, (short)0, c, false, false);
  }

  // C/D layout: lane l, VGPR j -> row m0 + j + 8*half, col n0 + l16
  float* o = T + (size_t)(m0 + half * 8) * DF + n0 + l16;
#pragma unroll
  for (int j = 0; j < 8; ++j) {
    if (m0 + half * 8 + j < nrows) o[(size_t)j * DF] = c[j];
  }
}

// ---------------- per-layer epilogue kernels ----------------
// Y = dinv[node]^2 * T + bias   (self-loop term + bias, also zero-initializes Y)
__global__ void k_init_out(const float* __restrict__ T, const float* __restrict__ dinv,
                           const float* __restrict__ bias, float* __restrict__ Y, int n) {
  int i = blockIdx.x * blockDim.x + threadIdx.x;
  if (i < n) {
    int node = i >> 7;
    int f = i & (DF - 1);
    float d = dinv[node];
    Y[i] = d * d * T[i] + bias[f];
  }
}

// Edge scatter: one wave per edge; each lane handles 4 features (float4 gather + 4 atomic adds).
__global__ void k_scatter(const long long* __restrict__ ei, const float* __restrict__ norm,
                          const float* __restrict__ T, float* __restrict__ Y, int E) {
  int t = blockIdx.x * blockDim.x + threadIdx.x;
  int e = t >> 5;
  int lane = t & 31;
  if (e >= E) return;
  int src = (int)ei[e];
  int dst = (int)ei[(size_t)E + e];
  float w = norm[e];
  const float4 v = *(const float4*)(T + (size_t)src * DF + lane * 4);
  float* yp = Y + (size_t)dst * DF + lane * 4;
  atomicAdd(yp + 0, w * v.x);
  atomicAdd(yp + 1, w * v.y);
  atomicAdd(yp + 2, w * v.z);
  atomicAdd(yp + 3, w * v.w);
}

__global__ void k_elu(float* Y, int n) {
  int i = blockIdx.x * blockDim.x + threadIdx.x;
  if (i < n) {
    float v = Y[i];
    Y[i] = v > 0.0f ? v : expm1f(v);
  }
}

// ---------------- pooling + readout MLP ----------------
__global__ void k_pool_zero(float* sums, float* cnt, int G) {
  int i = blockIdx.x * blockDim.x + threadIdx.x;
  if (i < G * DF) sums[i] = 0.0f;
  if (i < G) cnt[i] = 0.0f;
}

__global__ void k_pool(const float* __restrict__ H, const long long* __restrict__ batch,
                       float* __restrict__ sums, float* __restrict__ cnt, int n) {
  int i = blockIdx.x * blockDim.x + threadIdx.x;
  if (i < n) {
    int node = i >> 7;
    int f = i & (DF - 1);
    int g = (int)batch[node];
    atomicAdd(&sums[(size_t)g * DF + f], H[i]);
    if (f == 0) atomicAdd(&cnt[g], 1.0f);
  }
}

// One block (128 threads) per graph; tiny MLP 128 -> 64 -> 32 -> 1.
__global__ void k_mlp(const float* __restrict__ sums, const float* __restrict__ cnt,
                      const float* __restrict__ fw1, const float* __restrict__ fb1,
                      const float* __restrict__ fw2, const float* __restrict__ fb2,
                      const float* __restrict__ fw3, const float* __restrict__ fb3,
                      float* __restrict__ out) {
  __shared__ float g[DF];
  __shared__ float g1[64];
  __shared__ float g2[32];
  int gid = blockIdx.x;
  int t = threadIdx.x;
  float c = fmaxf(cnt[gid], 1.0f);
  g[t] = sums[(size_t)gid * DF + t] / c;
  __syncthreads();
  if (t < 64) {
    float acc = fb1[t];
    for (int f = 0; f < DF; ++f) acc += g[f] * fw1[(size_t)f * 64 + t];
    g1[t] = fmaxf(acc, 0.0f);
  }
  __syncthreads();
  if (t < 32) {
    float acc = fb2[t];
    for (int j = 0; j < 64; ++j) acc += g1[j] * fw2[(size_t)j * 32 + t];
    g2[t] = fmaxf(acc, 0.0f);
  }
  __syncthreads();
  if (t == 0) {
    float acc = fb3[0];
    for (int j = 0; j < 32; ++j) acc += g2[j] * fw3[j];
    out[gid] = acc;
  }
}

// ---------------- host launcher ----------------
extern "C" void kernel_launch(void* const* d_in, const int* in_sizes, int n_in,
                              void* d_out, int out_size, void* d_ws, size_t ws_size,
                              hipStream_t stream) {
  const float* x   = (const float*)d_in[0];
  const float* W1  = (const float*)d_in[1];
  const float* b1  = (const float*)d_in[2];
  const float* W2  = (const float*)d_in[3];
  const float* b2  = (const float*)d_in[4];
  const float* W3  = (const float*)d_in[5];
  const float* b3  = (const float*)d_in[6];
  const float* fw1 = (const float*)d_in[7];
  const float* fb1 = (const float*)d_in[8];
  const float* fw2 = (const float*)d_in[9];
  const float* fb2 = (const float*)d_in[10];
  const float* fw3 = (const float*)d_in[11];
  const float* fb3 = (const float*)d_in[12];
  const long long* ei    = (const long long*)d_in[13];
  const long long* batch = (const long long*)d_in[14];
  float* out = (float*)d_out;

  const int N = in_sizes[0] / DF;   // 50000
  const int E = in_sizes[13] / 2;   // 800000
  const int G = out_size;           // 64

  // workspace layout (256B aligned chunks)
  char* ws = (char*)d_ws;
  size_t off = 0;
  auto alloc = [&](size_t bytes) -> float* {
    float* p = (float*)(ws + off);
    off += (bytes + 255) & ~(size_t)255;
    return p;
  };
  float* T    = alloc((size_t)N * DF * 4);  // GEMM output (messages before scatter)
  float* H    = alloc((size_t)N * DF * 4);  // layer output / next layer input
  float* dinv = alloc((size_t)N * 4);       // degree -> rsqrt(degree)
  float* norm = alloc((size_t)E * 4);       // per-edge symmetric norm
  float* sums = alloc((size_t)G * DF * 4);
  float* cnt  = alloc((size_t)G * 4);

  const int TB = 256;
  dim3 blk(TB);
  int nb_nodes  = (N + TB - 1) / TB;
  int nb_edges  = (E + TB - 1) / TB;
  int nb_feat   = (N * DF + TB - 1) / TB;
  int nb_scat   = (int)(((size_t)E * 32 + TB - 1) / TB);
  int nb_gemm   = (N + 15) / 16;

  // degree + norm (computed once, reused for all 3 layers)
  k_deg_init<<<nb_nodes, blk, 0, stream>>>(dinv, N);
  k_deg_count<<<nb_edges, blk, 0, stream>>>(ei, dinv, E);
  k_dinv<<<nb_nodes, blk, 0, stream>>>(dinv, N);
  k_norm<<<nb_edges, blk, 0, stream>>>(ei, dinv, norm, E);

  // layer 1: x -> H (ELU)
  k_gemm<<<nb_gemm, blk, 0, stream>>>(x, W1, T, N);
  k_init_out<<<nb_feat, blk, 0, stream>>>(T, dinv, b1, H, N * DF);
  k_scatter<<<nb_scat, blk, 0, stream>>>(ei, norm, T, H, E);
  k_elu<<<nb_feat, blk, 0, stream>>>(H, N * DF);

  // layer 2: H -> H (ELU)
  k_gemm<<<nb_gemm, blk, 0, stream>>>(H, W2, T, N);
  k_init_out<<<nb_feat, blk, 0, stream>>>(T, dinv, b2, H, N * DF);
  k_scatter<<<nb_scat, blk, 0, stream>>>(ei, norm, T, H, E);
  k_elu<<<nb_feat, blk, 0, stream>>>(H, N * DF);

  // layer 3: H -> H (no activation)
  k_gemm<<<nb_gemm, blk, 0, stream>>>(H, W3, T, N);
  k_init_out<<<nb_feat, blk, 0, stream>>>(T, dinv, b3, H, N * DF);
  k_scatter<<<nb_scat, blk, 0, stream>>>(ei, norm, T, H, E);

  // global mean pool + readout MLP
  k_pool_zero<<<(G * DF + TB - 1) / TB, blk, 0, stream>>>(sums, cnt, G);
  k_pool<<<nb_feat, blk, 0, stream>>>(H, batch, sums, cnt, N * DF);
  k_mlp<<<G, dim3(DF), 0, stream>>>(sums, cnt, fw1, fb1, fw2, fb2, fw3, fb3, out);
}